// DSPFeatureExtractor_53145925321330
// MI455X (gfx1250) — compile-verified
//
#include <hip/hip_runtime.h>
#include <hip/hip_bf16.h>
#include <math.h>

// ---------------------------------------------------------------------------
// DSP feature extractor for MI455X (gfx1250, wave32, WMMA).
//  x: (131072, 1000) fp32  ->  out: (131072, 18) fp32
//
// Spectral part = DFT as GEMM on V_WMMA_F32_16X16X4_F32 (fp32 matrix op).
// Twiddle B-matrix pre-swizzled into per-lane WMMA fragment layout (4 MB ws,
// L2 resident). Each 128-thread WG owns a 64-row strip staged once in LDS;
// wave w computes M-tile pair (w>>1) over N-tile half (w&1): every B fragment
// feeds 4 WMMAs (2 M-tiles x re/im), halving L2 twiddle traffic vs 1 M-tile
// per wave. Spectral stats reduce in registers, then a small LDS combine
// merges the two N-halves per row.
// ---------------------------------------------------------------------------

#define L_IN        1000
#define NBINS       501          // rfft bins
#define NTILES      32           // 32 * 16 = 512 padded bins
#define KSTEPS      250          // 1000 / 4
#define BSTRIDE     (NTILES * 32) // v2f elements per ktile in frag layout
#define ROWS_PER_WG 64
#define LDS_STRIDE  1004         // 1004 dwords = 251 x 16 B; 251 coprime with 64 banks
#define NFEAT       18
#define EPS         1e-8f

typedef float  v2f __attribute__((ext_vector_type(2)));
typedef float  v8f __attribute__((ext_vector_type(8)));

// ---------------------------------------------------------------------------
// Kernel 0: twiddle fragments in the per-lane layout of the 4x16 fp32 WMMA
// B operand: frag[ktile][ntile][lane] = { W[k0+koff][n], W[k0+koff+1][n] },
// koff = (lane>=16) ? 2 : 0, n = ntile*16 + (lane&15).
// Re = cos(2*pi*t*n/1000), Im = -sin(2*pi*t*n/1000). Columns n>500 are zero.
// ---------------------------------------------------------------------------
__global__ void dsp_twiddle_init(float* __restrict__ bre, float* __restrict__ bim) {
  int idx = blockIdx.x * blockDim.x + threadIdx.x;     // over 250*32*32 entries
  if (idx >= KSTEPS * NTILES * 32) return;
  int lane  = idx & 31;
  int ntile = (idx >> 5) & 31;
  int ktile = idx >> 10;
  int n     = ntile * 16 + (lane & 15);
  int kbase = ktile * 4 + ((lane >> 4) << 1);
#pragma unroll
  for (int e = 0; e < 2; ++e) {
    int   t  = kbase + e;                              // time sample index
    float re = 0.f, im = 0.f;
    if (n <= NBINS - 1) {
      int   m   = (t * n) % L_IN;                      // < 500*999, fits int32
      float ang = -6.28318530717958647692f * 0.001f * (float)m;
      float s, c;
      sincosf(ang, &s, &c);
      re = c;
      im = s;
    }
    bre[idx * 2 + e] = re;
    bim[idx * 2 + e] = im;
  }
}

// ---------------------------------------------------------------------------
// Kernel 1: time-domain stats, one 256-thread block per row (memory bound).
// ---------------------------------------------------------------------------
__device__ __forceinline__ float fsign(float a) {
  return (a > 0.f) ? 1.f : ((a < 0.f) ? -1.f : 0.f);
}

__global__ __launch_bounds__(256)
void dsp_time_stats(const float* __restrict__ x, float* __restrict__ out) {
  const int   L   = L_IN;
  const int   row = blockIdx.x;
  const float* xr = x + (size_t)row * L;
  const int    t  = threadIdx.x;

  float s1 = 0.f, s2 = 0.f, s3 = 0.f, s4 = 0.f;
  float sd1sq = 0.f, sd2sq = 0.f, zc = 0.f;
  float mn = 3.4e38f, mx = -3.4e38f;

  int i0 = t * 4;                                      // threads 0..249 active
  if (i0 < L) {
    float v[6];
#pragma unroll
    for (int j = 0; j < 6; ++j) {
      int i = i0 + j;
      v[j] = (i < L) ? xr[i] : 0.f;
    }
#pragma unroll
    for (int j = 0; j < 4; ++j) {
      float a  = v[j];
      float a2 = a * a;
      s1 += a;  s2 += a2;  s3 += a2 * a;  s4 += a2 * a2;
      mn = fminf(mn, a);  mx = fmaxf(mx, a);
      if (i0 + j + 1 < L) {
        float d = v[j + 1] - v[j];
        sd1sq += d * d;
        if (fsign(v[j + 1]) != fsign(v[j])) zc += 1.f;
      }
      if (i0 + j + 2 < L) {
        float dd = v[j + 2] - 2.f * v[j + 1] + v[j];
        sd2sq += dd * dd;
      }
    }
  }

  __shared__ float red[9][256];
  red[0][t] = s1;    red[1][t] = s2;    red[2][t] = s3;   red[3][t] = s4;
  red[4][t] = sd1sq; red[5][t] = sd2sq; red[6][t] = zc;
  red[7][t] = mn;    red[8][t] = mx;
  __syncthreads();
  for (int s = 128; s > 0; s >>= 1) {
    if (t < s) {
#pragma unroll
      for (int q = 0; q < 7; ++q) red[q][t] += red[q][t + s];
      red[7][t] = fminf(red[7][t], red[7][t + s]);
      red[8][t] = fmaxf(red[8][t], red[8][t + s]);
    }
    __syncthreads();
  }

  if (t == 0) {
    float S1 = red[0][0], S2 = red[1][0], S3 = red[2][0], S4 = red[3][0];
    float D1 = red[4][0], D2 = red[5][0], ZC = red[6][0];
    float MN = red[7][0], MX = red[8][0];
    float fL   = (float)L;
    float mean = S1 / fL;
    float varU = (S2 - S1 * S1 / fL) / (fL - 1.f);     // ddof=1
    float stdU = sqrtf(varU);
    float sumd1 = xr[L - 1] - xr[0];
    float sumd2 = (xr[L - 1] - xr[L - 2]) - (xr[1] - xr[0]);
    float n1 = fL - 1.f, n2 = fL - 2.f;
    float mean_slope = sumd1 / n1;
    float std_slope  = sqrtf((D1 - sumd1 * sumd1 / n1) / (n1 - 1.f));
    float mean_curv  = sumd2 / n2;
    float std_curv   = sqrtf((D2 - sumd2 * sumd2 / n2) / (n2 - 1.f));
    float m  = mean;
    float m3 = S3 / fL - 3.f * m * (S2 / fL) + 2.f * m * m * m;
    float m4 = S4 / fL - 4.f * m * (S3 / fL) + 6.f * m * m * (S2 / fL)
               - 3.f * m * m * m * m;
    float skew = m3 / (stdU * stdU * stdU + EPS);
    float kurt = m4 / (varU * varU + EPS);             // std^4 == varU^2

    float* o = out + (size_t)row * NFEAT;
    o[0]  = mean;
    o[1]  = stdU;
    o[2]  = varU;
    o[3]  = MX - MN;
    o[4]  = ZC / n1;
    o[5]  = S2 / fL;
    o[6]  = sqrtf(S2 / fL);
    o[7]  = mean_slope;
    o[8]  = std_slope;
    o[9]  = mean_curv;
    o[10] = std_curv;
    o[11] = skew;
    o[12] = kurt;
  }
}

// ---------------------------------------------------------------------------
// Kernel 2: spectral features via fp32 WMMA DFT.
// ---------------------------------------------------------------------------
__global__ __launch_bounds__(128)
void dsp_spectral_wmma(const float* __restrict__ x,
                       const float* __restrict__ bre,
                       const float* __restrict__ bim,
                       float* __restrict__ out, int nrows) {
  extern __shared__ float lds[];                       // 64 * 1004 floats

  const int tid     = threadIdx.x;
  const int wave    = tid >> 5;
  const int lane    = tid & 31;
  const int halfsel = lane >> 4;                       // 0: lanes 0-15, 1: 16-31
  const int lrow    = lane & 15;
  const int mpair   = wave >> 1;                       // which 32-row half
  const int nhalf   = wave & 1;                        // which 16 N-tiles
  const int wgrow0  = blockIdx.x * ROWS_PER_WG;

  // stage the 64-row strip into LDS (clamp row for the ragged tail)
  for (int i = tid; i < ROWS_PER_WG * L_IN; i += 128) {
    int r  = i / L_IN;
    int c  = i - r * L_IN;
    int gr = wgrow0 + r;
    if (gr >= nrows) gr = nrows - 1;
    lds[r * LDS_STRIDE + c] = x[(size_t)gr * L_IN + c];
  }
  __syncthreads();

  // A fragment sources for the wave's two M-tiles (8B-aligned ds_load_b64)
  const float* aptr0 = lds + (mpair * 32 + lrow) * LDS_STRIDE + halfsel * 2;
  const float* aptr1 = aptr0 + 16 * LDS_STRIDE;

  // per-lane spectral accumulators; q = M-tile (0/1), VGPR r of D holds
  // local row q*16 + 8*halfsel + r, column n = ntile*16 + lrow
  float sumP[2][8], sumKP[2][8], sumK2P[2][8], hrP[2][8], rrP[2][8];
  float maxP[2][8], maxK[2][8];
#pragma unroll
  for (int q = 0; q < 2; ++q)
#pragma unroll
    for (int r = 0; r < 8; ++r) {
      sumP[q][r] = 0.f; sumKP[q][r] = 0.f; sumK2P[q][r] = 0.f;
      hrP[q][r] = 0.f;  rrP[q][r] = 0.f;
      maxP[q][r] = -1.f; maxK[q][r] = 0.f;
    }

  for (int nt = 0; nt < 16; ++nt) {
    const int ntile = nhalf * 16 + nt;
    const v2f* __restrict__ brp = (const v2f*)bre + (ntile * 32 + lane);
    const v2f* __restrict__ bip = (const v2f*)bim + (ntile * 32 + lane);
    v8f cre0 = {}, cim0 = {}, cre1 = {}, cim1 = {};
    v2f br = brp[0];
    v2f bi = bip[0];
    for (int kt = 0; kt < KSTEPS; ++kt) {
      // double-buffer: prefetch next ktile's B fragments before compute
      int kn = (kt < KSTEPS - 1) ? (kt + 1) : kt;
      v2f brn = brp[(size_t)kn * BSTRIDE];
      v2f bin = bip[(size_t)kn * BSTRIDE];
      v2f a0  = *(const v2f*)(aptr0 + kt * 4);
      v2f a1  = *(const v2f*)(aptr1 + kt * 4);
      cre0 = __builtin_amdgcn_wmma_f32_16x16x4_f32(
          false, a0, false, br, (short)0, cre0, false, false);
      cim0 = __builtin_amdgcn_wmma_f32_16x16x4_f32(
          false, a0, false, bi, (short)0, cim0, false, false);
      cre1 = __builtin_amdgcn_wmma_f32_16x16x4_f32(
          false, a1, false, br, (short)0, cre1, false, false);
      cim1 = __builtin_amdgcn_wmma_f32_16x16x4_f32(
          false, a1, false, bi, (short)0, cim1, false, false);
      br = brn;
      bi = bin;
    }
    // fold this 16-bin tile into the running per-row stats
    int   n     = ntile * 16 + lrow;
    bool  valid = (n <= NBINS - 1);
    float fk    = (float)n;
    float hrm   = (n >= 8 && n <= 30) ? 1.f : 0.f;     // 0.8..3.0 Hz
    float rrm   = (n >= 1 && n <= 5)  ? 1.f : 0.f;     // 0.1..0.5 Hz
#pragma unroll
    for (int q = 0; q < 2; ++q) {
#pragma unroll
      for (int r = 0; r < 8; ++r) {
        float re = (q == 0) ? cre0[r] : cre1[r];
        float im = (q == 0) ? cim0[r] : cim1[r];
        float P  = re * re + im * im;
        if (!valid) P = 0.f;
        sumP[q][r]   += P;
        sumKP[q][r]  += fk * P;
        sumK2P[q][r] += fk * fk * P;
        hrP[q][r]    += hrm * P;
        rrP[q][r]    += rrm * P;
        if (valid && P > maxP[q][r]) { maxP[q][r] = P; maxK[q][r] = fk; }
      }
    }
  }

  // intra-wave reduction across the 16 lanes of each half (wave32 shuffles)
#pragma unroll
  for (int q = 0; q < 2; ++q)
#pragma unroll
    for (int r = 0; r < 8; ++r)
#pragma unroll
      for (int m = 1; m <= 8; m <<= 1) {
        sumP[q][r]   += __shfl_xor(sumP[q][r],   m, 32);
        sumKP[q][r]  += __shfl_xor(sumKP[q][r],  m, 32);
        sumK2P[q][r] += __shfl_xor(sumK2P[q][r], m, 32);
        hrP[q][r]    += __shfl_xor(hrP[q][r],    m, 32);
        rrP[q][r]    += __shfl_xor(rrP[q][r],    m, 32);
        float oP = __shfl_xor(maxP[q][r], m, 32);
        float oK = __shfl_xor(maxK[q][r], m, 32);
        if (oP > maxP[q][r] || (oP == maxP[q][r] && oK < maxK[q][r])) {
          maxP[q][r] = oP; maxK[q][r] = oK;            // argmax -> first max
        }
      }

  // combine the two N-halves per row through LDS (reuse staging buffer)
  __syncthreads();                                     // all A reads done
  float* comb = lds;                                   // [wave][32 rows][8]
  if (lrow == 0) {
#pragma unroll
    for (int q = 0; q < 2; ++q)
#pragma unroll
      for (int r = 0; r < 8; ++r) {
        int    lr = q * 16 + halfsel * 8 + r;          // local row 0..31
        float* c  = comb + (((wave * 32) + lr) << 3);
        c[0] = sumP[q][r];  c[1] = sumKP[q][r]; c[2] = sumK2P[q][r];
        c[3] = hrP[q][r];   c[4] = rrP[q][r];
        c[5] = maxP[q][r];  c[6] = maxK[q][r];
      }
  }
  __syncthreads();

  if (tid < 64) {
    int mp = tid >> 5;                                 // 32-row half
    int lr = tid & 31;
    const float* pa = comb + ((((mp * 2 + 0) * 32) + lr) << 3);
    const float* pb = comb + ((((mp * 2 + 1) * 32) + lr) << 3);
    float tot = pa[0] + pb[0];
    float skp = pa[1] + pb[1];
    float sk2 = pa[2] + pb[2];
    float hr  = pa[3] + pb[3];
    float rr  = pa[4] + pb[4];
    float mP  = pa[5], mK = pa[6];
    if (pb[5] > mP || (pb[5] == mP && pb[6] < mK)) { mP = pb[5]; mK = pb[6]; }

    int row = wgrow0 + mp * 32 + lr;
    if (row < nrows) {
      float inv  = 1.f / (tot + EPS);
      float cent = 0.1f * skp * inv;
      float bw2  = (0.01f * sk2 - 0.2f * cent * skp + cent * cent * tot) * inv;
      float* o = out + (size_t)row * NFEAT;
      o[13] = 0.1f * mK;                               // dominant freq (Hz)
      o[14] = hr * inv;
      o[15] = rr * inv;
      o[16] = cent;
      o[17] = sqrtf(fmaxf(bw2, 0.f));
    }
  }
}

// ---------------------------------------------------------------------------
extern "C" void kernel_launch(void* const* d_in, const int* in_sizes, int n_in,
                              void* d_out, int out_size, void* d_ws, size_t ws_size,
                              hipStream_t stream) {
  const float* x   = (const float*)d_in[0];
  float*       out = (float*)d_out;
  const int    nrows = in_sizes[0] / L_IN;             // 131072

  // workspace: [bre: 250*32*32*2 floats][bim: same] = 4 MB, L2 resident
  float* bre = (float*)d_ws;
  float* bim = bre + KSTEPS * NTILES * 32 * 2;

  {
    int total  = KSTEPS * NTILES * 32;
    int blocks = (total + 255) / 256;
    dsp_twiddle_init<<<blocks, 256, 0, stream>>>(bre, bim);
  }

  dsp_time_stats<<<nrows, 256, 0, stream>>>(x, out);

  {
    int    blocks = (nrows + ROWS_PER_WG - 1) / ROWS_PER_WG;
    size_t smem   = (size_t)ROWS_PER_WG * LDS_STRIDE * sizeof(float);
    dsp_spectral_wmma<<<blocks, 128, smem, stream>>>(x, bre, bim, out, nrows);
  }
}